// Model_7722351199077
// MI455X (gfx1250) — compile-verified
//
#include <hip/hip_runtime.h>
#include <math.h>

// ---------------------------------------------------------------------------
// GPT-2 attention block on MI455X (gfx1250), fp32 end-to-end using
// V_WMMA_F32_16X16X4_F32 (wave32 matrix pipe, exact fp32 accumulate).
// Attention uses a flash-style online softmax; fully-unmasked k-tiles are
// processed in 32-column pairs so softmax bookkeeping (shuffles/exp/rescale)
// is amortized over 2x the columns; the causal diagonal tile is handled
// separately so the steady-state loop has no mask logic at all.
// ---------------------------------------------------------------------------

typedef float v2f __attribute__((ext_vector_type(2)));
typedef float v8f __attribute__((ext_vector_type(8)));

#define B_   4
#define S_   2048
#define D_   192
#define H_   8
#define HD_  24
#define ROWS (B_ * S_)                  // 8192
#define QKV_STRIDE (B_ * H_ * S_ * HD_) // floats per q/k/v tensor = 1572864

#define NEG_INF_MASK (-3.4028235e38f)
#define INV_SQRT_HD  (0.2041241452319315f)   // 1/sqrt(24)

// fp32 WMMA helper: D = A(16x4) * B(4x16) + C (one wave, wave32)
__device__ __forceinline__ v8f wmma_f32(v2f a, v2f b, v8f c) {
    return __builtin_amdgcn_wmma_f32_16x16x4_f32(
        /*neg_a=*/false, a, /*neg_b=*/false, b,
        /*c_mod=*/(short)0, c, /*reuse_a=*/false, /*reuse_b=*/false);
}

// Butterfly reductions across the 16-lane half-wave group (xor masks 1..8
// stay inside each group of 16 lanes on wave32).
__device__ __forceinline__ float rowMax16(float v) {
    v = fmaxf(v, __shfl_xor(v, 1, 32));
    v = fmaxf(v, __shfl_xor(v, 2, 32));
    v = fmaxf(v, __shfl_xor(v, 4, 32));
    v = fmaxf(v, __shfl_xor(v, 8, 32));
    return v;
}
__device__ __forceinline__ float rowSum16(float v) {
    v += __shfl_xor(v, 1, 32);
    v += __shfl_xor(v, 2, 32);
    v += __shfl_xor(v, 4, 32);
    v += __shfl_xor(v, 8, 32);
    return v;
}

// ---------------------------------------------------------------------------
// Kernel 1: qkv = x @ w_attn + b_attn, scattered to [3][B*H][S][HD] layout.
// One wave per 16x16 output tile. M = 8192, N = 576, K = 192.
// ---------------------------------------------------------------------------
__global__ __launch_bounds__(128)
void qkv_kernel(const float* __restrict__ x, const float* __restrict__ w,
                const float* __restrict__ bias, float* __restrict__ qkv_ws) {
    const int n_tiles = (3 * D_) / 16;                    // 36
    int wave = blockIdx.x * (blockDim.x >> 5) + (threadIdx.x >> 5);
    int mt = wave / n_tiles, nt = wave % n_tiles;
    int m0 = mt * 16, n0 = nt * 16;
    int lane = threadIdx.x & 31, half = lane >> 4, l16 = lane & 15;

    v8f acc = {};
    const float* arow = x + (size_t)(m0 + l16) * D_;      // lanes 0-15/16-31 both map M=l16
    #pragma unroll 4
    for (int k0 = 0; k0 < D_; k0 += 4) {
        int kb = k0 + half * 2;                           // fp32 A layout: K pair per half
        v2f a = *(const v2f*)(arow + kb);                 // contiguous, 8B aligned
        v2f b;
        b.x = w[(size_t)kb * (3 * D_) + n0 + l16];
        b.y = w[(size_t)(kb + 1) * (3 * D_) + n0 + l16];
        acc = wmma_f32(a, b, acc);
    }

    int n = n0 + l16;
    int which = n / D_, d = n % D_, h = d / HD_, hd = d % HD_;
    float bn = bias[n];
    int bidx = m0 / S_, sbase = m0 % S_;
    float* outp = qkv_ws + (size_t)which * QKV_STRIDE
                + ((size_t)(bidx * H_ + h) * S_ + sbase) * HD_ + hd;
    #pragma unroll
    for (int v = 0; v < 8; ++v) {
        int r = v + 8 * half;                             // D layout: row = v + 8*(lane/16)
        outp[(size_t)r * HD_] = acc[v] + bn;
    }
}

// ---------------------------------------------------------------------------
// Kernel 2: causal flash attention, one wave per (b, h, 16-row q-tile).
// ---------------------------------------------------------------------------
#define PSTRIDE 33   // 16x32 P tile per wave; 33*l16 mod 64 conflict-free

// Scores tile: S16x16 = Q(16x24) * K^T(24x16), padded K-loop of 6 fp32 WMMAs.
__device__ __forceinline__ v8f score_tile(const float* __restrict__ qrow,
                                          const float* __restrict__ kp,
                                          int kv0, int l16, int half) {
    v8f sacc = {};
    const float* krow = kp + (size_t)(kv0 + l16) * HD_;   // K^T col n=l16 = K row kv0+l16
    #pragma unroll
    for (int kk = 0; kk < HD_; kk += 4) {
        int kb = kk + half * 2;
        v2f a  = *(const v2f*)(qrow + kb);                // loop-invariant, hoisted
        v2f bb = *(const v2f*)(krow + kb);
        sacc = wmma_f32(a, bb, sacc);
    }
    return sacc;
}

// O += P(16x16) @ V(16x24): 4 K-steps x 2 N-halves; P read from LDS in A-layout.
__device__ __forceinline__ void pv16(const float* __restrict__ sm,
                                     const float* __restrict__ vp,
                                     int kv0, int cbase, int l16, int half,
                                     v8f& o_lo, v8f& o_hi) {
    #pragma unroll
    for (int kk = 0; kk < 16; kk += 4) {
        int kb = kk + half * 2;
        v2f a;
        a.x = sm[l16 * PSTRIDE + cbase + kb];             // A layout: M=l16, K=kb
        a.y = sm[l16 * PSTRIDE + cbase + kb + 1];
        const float* vrow0 = vp + (size_t)(kv0 + kb) * HD_;
        const float* vrow1 = vrow0 + HD_;
        v2f blo; blo.x = vrow0[l16]; blo.y = vrow1[l16];
        o_lo = wmma_f32(a, blo, o_lo);
        int hd2 = 16 + l16;
        v2f bhi;
        bhi.x = (hd2 < HD_) ? vrow0[hd2] : 0.0f;          // pad hd 24..31 with zeros
        bhi.y = (hd2 < HD_) ? vrow1[hd2] : 0.0f;
        o_hi = wmma_f32(a, bhi, o_hi);
    }
}

__global__ __launch_bounds__(128)
void attn_kernel(const float* __restrict__ q_ws, const float* __restrict__ k_ws,
                 const float* __restrict__ v_ws, const float* __restrict__ mask0,
                 float* __restrict__ attn_ws) {
    __shared__ float smem[4 * 16 * PSTRIDE];              // 4 waves/block, private regions

    int wave_in_blk = threadIdx.x >> 5;
    int wid = blockIdx.x * 4 + wave_in_blk;
    int qt = wid & 127;                                   // S/16 = 128 q-tiles
    int bh = wid >> 7;                                    // 0..31
    int h = bh & 7, b = bh >> 3;
    int lane = threadIdx.x & 31, half = lane >> 4, l16 = lane & 15;

    float* sm = smem + wave_in_blk * 16 * PSTRIDE;
    const float* qp = q_ws + (size_t)bh * S_ * HD_;
    const float* kp = k_ws + (size_t)bh * S_ * HD_;
    const float* vp = v_ws + (size_t)bh * S_ * HD_;
    const float* mp = mask0 + (size_t)b * S_;

    int q0 = qt * 16;
    v8f o_lo = {};                                        // hd 0..15
    v8f o_hi = {};                                        // hd 16..23 (cols 24..31 dead)
    float rmax[8], rsum[8];
    #pragma unroll
    for (int v = 0; v < 8; ++v) { rmax[v] = NEG_INF_MASK; rsum[v] = 0.0f; }

    const float* qrow = qp + (size_t)(q0 + l16) * HD_;

    // ---- steady state: fully-unmasked k-tiles (j < qt) in 32-column pairs.
    // One softmax update (one butterfly max, one butterfly sum, one rescale)
    // covers 2 tiles -> 28 WMMAs per update.
    int npairs = qt >> 1;
    for (int p = 0; p < npairs; ++p) {
        int kv0 = p * 32;
        v8f s0 = score_tile(qrow, kp, kv0,      l16, half);
        v8f s1 = score_tile(qrow, kp, kv0 + 16, l16, half);
        float madd0 = mp[kv0 + l16];
        float madd1 = mp[kv0 + 16 + l16];
        #pragma unroll
        for (int v = 0; v < 8; ++v) {
            float a0 = s0[v] * INV_SQRT_HD + madd0;       // no causal mask below diagonal
            float a1 = s1[v] * INV_SQRT_HD + madd1;
            float tmax = rowMax16(fmaxf(a0, a1));
            float mnew = fmaxf(rmax[v], tmax);
            float p0 = __expf(a0 - mnew);
            float p1 = __expf(a1 - mnew);
            float tsum = rowSum16(p0 + p1);
            float so = __expf(rmax[v] - mnew);            // rescale old state once per 32 cols
            rsum[v] = rsum[v] * so + tsum;
            rmax[v] = mnew;
            o_lo[v] *= so;
            o_hi[v] *= so;
            int r = v + 8 * half;
            sm[r * PSTRIDE + l16]      = p0;              // stage 16x32 P stripe
            sm[r * PSTRIDE + 16 + l16] = p1;
        }
        pv16(sm, vp, kv0,      0,  l16, half, o_lo, o_hi);
        pv16(sm, vp, kv0 + 16, 16, l16, half, o_lo, o_hi);
    }

    // ---- tail: leftover unmasked tile (if qt odd) + causal diagonal tile.
    for (int j = 2 * npairs; j <= qt; ++j) {
        int kv0 = j * 16;
        v8f sacc = score_tile(qrow, kp, kv0, l16, half);
        int colk = kv0 + l16;
        float madd = mp[colk];
        bool diag = (j == qt);
        #pragma unroll
        for (int v = 0; v < 8; ++v) {
            int rowq = q0 + v + 8 * half;
            float s = sacc[v] * INV_SQRT_HD;
            if (diag && colk > rowq) s = NEG_INF_MASK;    // masked_fill with dtype-min
            s += madd;                                    // additive HF padding mask
            float tmax = rowMax16(s);
            float mnew = fmaxf(rmax[v], tmax);
            float pe   = __expf(s - mnew);
            float tsum = rowSum16(pe);
            float so   = __expf(rmax[v] - mnew);
            rsum[v] = rsum[v] * so + tsum;
            rmax[v] = mnew;
            o_lo[v] *= so;
            o_hi[v] *= so;
            sm[(v + 8 * half) * PSTRIDE + l16] = pe;
        }
        pv16(sm, vp, kv0, 0, l16, half, o_lo, o_hi);
    }

    // ---- normalize and store into [B*S, 192] row-major (pre-projection)
    int colbase = h * HD_;
    #pragma unroll
    for (int v = 0; v < 8; ++v) {
        float inv = 1.0f / rsum[v];
        int srow = q0 + v + 8 * half;
        size_t rowoff = ((size_t)b * S_ + srow) * D_ + colbase;
        attn_ws[rowoff + l16] = o_lo[v] * inv;
        if (l16 < 8) attn_ws[rowoff + 16 + l16] = o_hi[v] * inv;
    }
}

// ---------------------------------------------------------------------------
// Kernel 3: out = attn @ w_proj + b_proj. M = 8192, N = 192, K = 192.
// ---------------------------------------------------------------------------
__global__ __launch_bounds__(128)
void proj_kernel(const float* __restrict__ a_ws, const float* __restrict__ w,
                 const float* __restrict__ bias, float* __restrict__ out) {
    const int n_tiles = D_ / 16;                          // 12
    int wave = blockIdx.x * (blockDim.x >> 5) + (threadIdx.x >> 5);
    int mt = wave / n_tiles, nt = wave % n_tiles;
    int m0 = mt * 16, n0 = nt * 16;
    int lane = threadIdx.x & 31, half = lane >> 4, l16 = lane & 15;

    v8f acc = {};
    const float* arow = a_ws + (size_t)(m0 + l16) * D_;
    #pragma unroll 4
    for (int k0 = 0; k0 < D_; k0 += 4) {
        int kb = k0 + half * 2;
        v2f a = *(const v2f*)(arow + kb);
        v2f b;
        b.x = w[(size_t)kb * D_ + n0 + l16];
        b.y = w[(size_t)(kb + 1) * D_ + n0 + l16];
        acc = wmma_f32(a, b, acc);
    }

    int n = n0 + l16;
    float bn = bias[n];
    #pragma unroll
    for (int v = 0; v < 8; ++v) {
        int r = m0 + v + 8 * half;
        out[(size_t)r * D_ + n] = acc[v] + bn;
    }
}

// ---------------------------------------------------------------------------
extern "C" void kernel_launch(void* const* d_in, const int* in_sizes, int n_in,
                              void* d_out, int out_size, void* d_ws, size_t ws_size,
                              hipStream_t stream) {
    const float* x      = (const float*)d_in[0];
    const float* mask0  = (const float*)d_in[1];
    const float* w_attn = (const float*)d_in[2];
    const float* b_attn = (const float*)d_in[3];
    const float* w_proj = (const float*)d_in[4];
    const float* b_proj = (const float*)d_in[5];
    float* out = (float*)d_out;
    float* wsf = (float*)d_ws;

    float* q_ws    = wsf;                           // [B*H][S][24]
    float* k_ws    = wsf + (size_t)QKV_STRIDE;
    float* v_ws    = wsf + (size_t)2 * QKV_STRIDE;
    float* attn_ws = wsf + (size_t)3 * QKV_STRIDE;  // [B*S][192]

    // 1) QKV projection: 512 * 36 tiles, 1 wave/tile, 4 waves/block
    {
        int waves = (ROWS / 16) * ((3 * D_) / 16);   // 18432
        qkv_kernel<<<waves / 4, 128, 0, stream>>>(x, w_attn, b_attn, wsf);
    }
    // 2) causal attention: B*H*(S/16) = 4096 waves
    {
        int waves = B_ * H_ * (S_ / 16);
        attn_kernel<<<waves / 4, 128, 0, stream>>>(q_ws, k_ws, v_ws, mask0, attn_ws);
    }
    // 3) output projection: 512 * 12 tiles
    {
        int waves = (ROWS / 16) * (D_ / 16);         // 6144
        proj_kernel<<<waves / 4, 128, 0, stream>>>(attn_ws, w_proj, b_proj, out);
    }
}